// Grapher_61787399520597
// MI455X (gfx1250) — compile-verified
//
#include <hip/hip_runtime.h>
#include <hip/hip_bf16.h>
#include <math.h>

// ---------------- problem constants ----------------
#define Bdim 8
#define Cdim 192
#define Hdim 56
#define Wdim 56
#define HWdim (Hdim * Wdim)          // 3136
#define NELEM ((size_t)Bdim * Cdim * HWdim)  // 4,816,896
#define NSHIFT 13                    // shifts 4,8,...,52
#define BN_EPS 1e-5f

typedef __attribute__((ext_vector_type(2))) float v2f;
typedef __attribute__((ext_vector_type(8))) float v8f;

// ---------------- BN folding: weff = w*scale, beff = bias*scale + beta - m*scale ----
__global__ void fold_bn_kernel(const float* __restrict__ w, const float* __restrict__ bias,
                               const float* __restrict__ g, const float* __restrict__ beta,
                               const float* __restrict__ m, const float* __restrict__ v,
                               float* __restrict__ weff, float* __restrict__ beff,
                               int M, int K) {
    int idx = blockIdx.x * blockDim.x + threadIdx.x;
    if (idx >= M * K) return;
    int o = idx / K;
    int k = idx - o * K;
    float scale = g[o] / sqrtf(v[o] + BN_EPS);
    weff[idx] = w[idx] * scale;
    if (k == 0) beff[o] = bias[o] * scale + beta[o] - m[o] * scale;
}

// ---------------- depthwise 7x7 conv + bias + residual ----------------
__global__ __launch_bounds__(256) void dwconv7_kernel(const float* __restrict__ x,
                                                      const float* __restrict__ pw,
                                                      const float* __restrict__ pb,
                                                      float* __restrict__ out) {
    size_t idx = (size_t)blockIdx.x * blockDim.x + threadIdx.x;
    if (idx >= NELEM) return;
    int w = (int)(idx % Wdim);
    size_t t = idx / Wdim;
    int h = (int)(t % Hdim); t /= Hdim;
    int c = (int)(t % Cdim);
    int b = (int)(t / Cdim);
    const float* xb = x + ((size_t)(b * Cdim + c)) * HWdim;
    const float* wc = pw + c * 49;
    float acc = pb[c];
#pragma unroll
    for (int i = 0; i < 7; ++i) {
        int hh = h + i - 3;
        if ((unsigned)hh >= (unsigned)Hdim) continue;
#pragma unroll
        for (int j = 0; j < 7; ++j) {
            int ww = w + j - 3;
            if ((unsigned)ww >= (unsigned)Wdim) continue;
            acc = fmaf(wc[i * 7 + j], xb[hh * Wdim + ww], acc);
        }
    }
    out[idx] = acc + xb[h * Wdim + w];
}

// ---------------- WMMA fp32 GEMM: Out[b][m][n] = sum_k Weff[m][k] * X[b][k][n] + beff[m]
// Weight columns [0,K-K2) read activations from X1, [K-K2,K) from X2 (concat without
// materialization). Split K loops keep the steady state branch-free.
// One wave computes 4 M-tiles x 2 N-tiles (16x16 each), K-step 4 via V_WMMA_F32_16X16X4_F32.
// Operand layouts per CDNA5 ISA 7.12.2:
//   A 16x4 : M = lane&15 ; VGPR{0,1} = K {0,1} (lanes 0-15) / K {2,3} (lanes 16-31)
//   B 4x16 : N = lane&15 ; VGPR{0,1} = K {0,1} (lanes 0-15) / K {2,3} (lanes 16-31)
//   C/D    : N = lane&15 ; M = 8*(lane>>4) + vgpr
__global__ __launch_bounds__(256) void gemm_wmma_kernel(const float* __restrict__ X1,
                                                        const float* __restrict__ X2,
                                                        const float* __restrict__ Weff,
                                                        const float* __restrict__ beff,
                                                        float* __restrict__ Out,
                                                        int K, int K2, int act_gelu) {
    const int Npairs = HWdim / 32;        // 98 (2 N-tiles per wave)
    const int ngroups = (Cdim / 16) / 4;  // 12 M-tiles, 4 per wave -> 3 groups
    int wid = blockIdx.x * (blockDim.x >> 5) + (threadIdx.x >> 5);
    int lane = threadIdx.x & 31;
    int total = Bdim * ngroups * Npairs;  // 2352 waves, grid sized exactly
    if (wid >= total) return;

    int np = wid % Npairs;
    int t = wid / Npairs;
    int mg = t % ngroups;
    int b = t / ngroups;

    int koff = (lane & 16) ? 2 : 0;
    int l15 = lane & 15;
    int n0 = np * 32 + l15;
    int n1 = n0 + 16;
    int m0 = mg * 64;                     // first output channel of this wave's 4 M-tiles
    int mrow = (lane >> 4) * 8;
    int K1 = K - K2;

    const float* xb1 = X1 + (size_t)b * Cdim * HWdim;
    const float* xb2 = X2 + (size_t)b * Cdim * HWdim;
    const float* wrow = Weff + (size_t)(m0 + l15) * K;  // per-lane A row base

    // init accumulators with the folded bias (C-matrix = bias broadcast over N)
    v8f acc[4][2];
#pragma unroll
    for (int i = 0; i < 4; ++i) {
        v8f a;
#pragma unroll
        for (int vv = 0; vv < 8; ++vv) a[vv] = beff[m0 + i * 16 + mrow + vv];
        acc[i][0] = a;
        acc[i][1] = a;
    }

    // ---- K segment 1: activations from X1 ----
    for (int k0 = 0; k0 < K1; k0 += 4) {
        int k = k0 + koff;
        const float* src = xb1 + (size_t)k * HWdim;
        v2f bv0, bv1;
        bv0.x = src[n0]; bv0.y = src[HWdim + n0];
        bv1.x = src[n1]; bv1.y = src[HWdim + n1];
#pragma unroll
        for (int i = 0; i < 4; ++i) {
            v2f av = *(const v2f*)(wrow + (size_t)i * 16 * K + k);
            acc[i][0] = __builtin_amdgcn_wmma_f32_16x16x4_f32(false, av, false, bv0, (short)0, acc[i][0], false, false);
            acc[i][1] = __builtin_amdgcn_wmma_f32_16x16x4_f32(false, av, false, bv1, (short)0, acc[i][1], false, false);
        }
    }

    // ---- K segment 2: activations from X2 (concat channels) ----
    for (int k0 = K1; k0 < K; k0 += 4) {
        int k = k0 + koff;
        const float* src = xb2 + (size_t)(k - K1) * HWdim;
        v2f bv0, bv1;
        bv0.x = src[n0]; bv0.y = src[HWdim + n0];
        bv1.x = src[n1]; bv1.y = src[HWdim + n1];
#pragma unroll
        for (int i = 0; i < 4; ++i) {
            v2f av = *(const v2f*)(wrow + (size_t)i * 16 * K + k);
            acc[i][0] = __builtin_amdgcn_wmma_f32_16x16x4_f32(false, av, false, bv0, (short)0, acc[i][0], false, false);
            acc[i][1] = __builtin_amdgcn_wmma_f32_16x16x4_f32(false, av, false, bv1, (short)0, acc[i][1], false, false);
        }
    }

    // ---- epilogue: optional exact GELU, store ----
    float* ob = Out + (size_t)b * Cdim * HWdim;
#pragma unroll
    for (int i = 0; i < 4; ++i) {
        int mt = m0 + i * 16 + mrow;
#pragma unroll
        for (int j = 0; j < 2; ++j) {
            int n = (j == 0) ? n0 : n1;
            v8f a = acc[i][j];
#pragma unroll
            for (int vv = 0; vv < 8; ++vv) {
                float val = a[vv];
                if (act_gelu) val = 0.5f * val * (1.0f + erff(val * 0.70710678118654752f));
                ob[(size_t)(mt + vv) * HWdim + n] = val;
            }
        }
    }
}

// ---------------- soft_weighted_diff along H (axis=2): one block per (b, w) column ----
__global__ __launch_bounds__(256) void sw_h_kernel(const float* __restrict__ x,
                                                   float* __restrict__ xh) {
    __shared__ float col[Cdim][Hdim];     // 43 KB
    __shared__ float sim[NSHIFT][Hdim];
    __shared__ float simsum[Hdim];
    int b = blockIdx.x / Wdim;
    int w = blockIdx.x % Wdim;
    const float* base = x + (size_t)b * Cdim * HWdim + w;
    for (int i = threadIdx.x; i < Cdim * Hdim; i += blockDim.x) {
        int c = i / Hdim, h = i - (i / Hdim) * Hdim;
        col[c][h] = base[(size_t)(c * Hdim + h) * Wdim];
    }
    __syncthreads();
    for (int p = threadIdx.x; p < NSHIFT * Hdim; p += blockDim.x) {
        int s = p / Hdim, h = p - s * Hdim;
        int hs = h - (s + 1) * 4; if (hs < 0) hs += Hdim;
        float acc = 0.f;
        for (int c = 0; c < Cdim; ++c) acc += fabsf(col[c][hs] - col[c][h]);
        sim[s][h] = expf(-acc * 10.0f);   // /TEMP, TEMP=0.1
    }
    __syncthreads();
    for (int h = threadIdx.x; h < Hdim; h += blockDim.x) {
        float ss = 1e-6f;
        for (int s = 0; s < NSHIFT; ++s) ss += sim[s][h];
        simsum[h] = ss;
    }
    __syncthreads();
    float* obase = xh + (size_t)b * Cdim * HWdim + w;
    for (int i = threadIdx.x; i < Cdim * Hdim; i += blockDim.x) {
        int c = i / Hdim, h = i - (i / Hdim) * Hdim;
        float xc = col[c][h];
        float num = 0.f;
#pragma unroll
        for (int s = 0; s < NSHIFT; ++s) {
            int hs = h - (s + 1) * 4; if (hs < 0) hs += Hdim;
            num += (col[c][hs] - xc) * sim[s][h];
        }
        obase[(size_t)(c * Hdim + h) * Wdim] = num / simsum[h];
    }
}

// ---------------- soft_weighted_diff along W (axis=3), fused max with xh -> xj ----------
__global__ __launch_bounds__(256) void sw_w_kernel(const float* __restrict__ x,
                                                   const float* __restrict__ xh,
                                                   float* __restrict__ xj) {
    __shared__ float row[Cdim][Wdim];
    __shared__ float sim[NSHIFT][Wdim];
    __shared__ float simsum[Wdim];
    int b = blockIdx.x / Hdim;
    int h = blockIdx.x % Hdim;
    const float* base = x + (size_t)b * Cdim * HWdim + (size_t)h * Wdim;
    for (int i = threadIdx.x; i < Cdim * Wdim; i += blockDim.x) {
        int c = i / Wdim, w = i - (i / Wdim) * Wdim;
        row[c][w] = base[(size_t)c * HWdim + w];
    }
    __syncthreads();
    for (int p = threadIdx.x; p < NSHIFT * Wdim; p += blockDim.x) {
        int s = p / Wdim, w = p - s * Wdim;
        int ws = w - (s + 1) * 4; if (ws < 0) ws += Wdim;
        float acc = 0.f;
        for (int c = 0; c < Cdim; ++c) acc += fabsf(row[c][ws] - row[c][w]);
        sim[s][w] = expf(-acc * 10.0f);
    }
    __syncthreads();
    for (int w = threadIdx.x; w < Wdim; w += blockDim.x) {
        float ss = 1e-6f;
        for (int s = 0; s < NSHIFT; ++s) ss += sim[s][w];
        simsum[w] = ss;
    }
    __syncthreads();
    const float* hbase = xh + (size_t)b * Cdim * HWdim + (size_t)h * Wdim;
    float* obase = xj + (size_t)b * Cdim * HWdim + (size_t)h * Wdim;
    for (int i = threadIdx.x; i < Cdim * Wdim; i += blockDim.x) {
        int c = i / Wdim, w = i - (i / Wdim) * Wdim;
        float xc = row[c][w];
        float num = 0.f;
#pragma unroll
        for (int s = 0; s < NSHIFT; ++s) {
            int ws = w - (s + 1) * 4; if (ws < 0) ws += Wdim;
            num += (row[c][ws] - xc) * sim[s][w];
        }
        float xw = num / simsum[w];
        float xhv = hbase[(size_t)c * HWdim + w];
        obase[(size_t)c * HWdim + w] = fmaxf(xhv, xw);
    }
}

// ---------------- launch ----------------
extern "C" void kernel_launch(void* const* d_in, const int* in_sizes, int n_in,
                              void* d_out, int out_size, void* d_ws, size_t ws_size,
                              hipStream_t stream) {
    const float* x     = (const float*)d_in[0];
    const float* pe_w  = (const float*)d_in[1];
    const float* pe_b  = (const float*)d_in[2];
    const float* fc1_w = (const float*)d_in[3];
    const float* fc1_b = (const float*)d_in[4];
    const float* fc1_g = (const float*)d_in[5];
    const float* fc1_be= (const float*)d_in[6];
    const float* fc1_m = (const float*)d_in[7];
    const float* fc1_v = (const float*)d_in[8];
    const float* gc_w  = (const float*)d_in[9];
    const float* gc_b  = (const float*)d_in[10];
    const float* gc_g  = (const float*)d_in[11];
    const float* gc_be = (const float*)d_in[12];
    const float* gc_m  = (const float*)d_in[13];
    const float* gc_v  = (const float*)d_in[14];
    const float* fc2_w = (const float*)d_in[15];
    const float* fc2_b = (const float*)d_in[16];
    const float* fc2_g = (const float*)d_in[17];
    const float* fc2_be= (const float*)d_in[18];
    const float* fc2_m = (const float*)d_in[19];
    const float* fc2_v = (const float*)d_in[20];
    float* out = (float*)d_out;

    float* ws   = (float*)d_ws;
    float* buf0 = ws;                 // y1 -> xh -> y4
    float* buf1 = buf0 + NELEM;       // y2 (fc1 output)
    float* buf2 = buf1 + NELEM;       // xj
    float* w1e  = buf2 + NELEM;
    float* b1e  = w1e + Cdim * Cdim;
    float* wge  = b1e + Cdim;
    float* bge  = wge + Cdim * 2 * Cdim;
    float* w2e  = bge + Cdim;
    float* b2e  = w2e + Cdim * Cdim;

    // 1) fold BN into weights
    fold_bn_kernel<<<(Cdim * Cdim + 255) / 256, 256, 0, stream>>>(
        fc1_w, fc1_b, fc1_g, fc1_be, fc1_m, fc1_v, w1e, b1e, Cdim, Cdim);
    fold_bn_kernel<<<(Cdim * 2 * Cdim + 255) / 256, 256, 0, stream>>>(
        gc_w, gc_b, gc_g, gc_be, gc_m, gc_v, wge, bge, Cdim, 2 * Cdim);
    fold_bn_kernel<<<(Cdim * Cdim + 255) / 256, 256, 0, stream>>>(
        fc2_w, fc2_b, fc2_g, fc2_be, fc2_m, fc2_v, w2e, b2e, Cdim, Cdim);

    // 2) depthwise 7x7 + residual : x -> buf0
    dwconv7_kernel<<<(unsigned)((NELEM + 255) / 256), 256, 0, stream>>>(x, pe_w, pe_b, buf0);

    // GEMM grid: Bdim * 3 mgroups * 98 npairs waves, 8 waves/block (exactly full waves)
    const int gemm_waves = Bdim * 3 * (HWdim / 32);   // 2352
    const int gemm_blocks = gemm_waves / 8;           // 294

    // 3) fc1 + BN : buf0 -> buf1
    gemm_wmma_kernel<<<gemm_blocks, 256, 0, stream>>>(buf0, buf0, w1e, b1e, buf1, Cdim, 0, 0);

    // 4) soft diff along H : buf1 -> buf0 (xh)
    sw_h_kernel<<<Bdim * Wdim, 256, 0, stream>>>(buf1, buf0);

    // 5) soft diff along W + max : (buf1, buf0) -> buf2 (xj)
    sw_w_kernel<<<Bdim * Hdim, 256, 0, stream>>>(buf1, buf0, buf2);

    // 6) gc (concat via dual input) + BN + GELU : (buf1 | buf2) -> buf0
    gemm_wmma_kernel<<<gemm_blocks, 256, 0, stream>>>(buf1, buf2, wge, bge, buf0, 2 * Cdim, Cdim, 1);

    // 7) fc2 + BN : buf0 -> d_out
    gemm_wmma_kernel<<<gemm_blocks, 256, 0, stream>>>(buf0, buf0, w2e, b2e, out, Cdim, 0, 0);
}